// Fuzzy_ElpiGraph_13305808683177
// MI455X (gfx1250) — compile-verified
//
#include <hip/hip_runtime.h>
#include <hip/hip_bf16.h>
#include <math.h>

typedef __attribute__((ext_vector_type(2))) float v2f;
typedef __attribute__((ext_vector_type(8))) float v8f;

#define N_X   100000
#define M_Y   1024
#define ALPHA 5.0f
#define LMDA  0.01f
#define MU    0.1f

// workspace layout (float offsets)
#define OFF_SQ      0                       // [1024*1024] raw pairwise sq on y
#define OFF_A       (1024*1024)             // [1024*1024] fuzzy graph A
#define OFF_ROWSUM  (2*1024*1024)           // [1024]
#define OFF_DEG     (OFF_ROWSUM + 1024)     // [1024]
#define OFF_AY      (OFF_DEG + 1024)        // [1024*64]
#define OFF_Y2      (OFF_AY + 1024*64)      // [1024]
#define OFF_X2      (OFF_Y2 + 1024)         // [100000]
#define OFF_ACC     (OFF_X2 + 100000)
// ACC: [0]=gmin_y bits(u32), [1]=gmin_X bits(u32), [2]=mse acc, [3]=stretch, [4]=harm

__device__ __forceinline__ v8f wmma_f32(v2f a, v2f b, v8f c) {
  // V_WMMA_F32_16X16X4_F32 : D = A(16x4) * B(4x16) + C
  return __builtin_amdgcn_wmma_f32_16x16x4_f32(
      /*neg_a=*/false, a, /*neg_b=*/false, b,
      /*c_mod=*/(short)0, c, /*reuse_a=*/false, /*reuse_b=*/false);
}

__global__ void k_init_acc(float* ws) {
  if (threadIdx.x == 0) {
    unsigned* u = (unsigned*)(ws + OFF_ACC);
    u[0] = 0x7F800000u;   // +inf
    u[1] = 0x7F800000u;
    ws[OFF_ACC + 2] = 0.f;
    ws[OFF_ACC + 3] = 0.f;
    ws[OFF_ACC + 4] = 0.f;
  }
}

// one wave per row, D=64 fixed: lane sums elem lane and lane+32
__global__ void k_rownorm(const float* __restrict__ p, float* __restrict__ out, int nrows) {
  int wave = threadIdx.x >> 5, lane = threadIdx.x & 31;
  int row = blockIdx.x * 8 + wave;
  if (row >= nrows) return;
  float a = p[row * 64 + lane];
  float b = p[row * 64 + lane + 32];
  float v = a * a + b * b;
  for (int m = 16; m; m >>= 1) v += __shfl_xor(v, m, 32);
  if (lane == 0) out[row] = v;
}

// y x y^T tiles: store sq, global min of (diag->100) matrix
__global__ void k_graph_dst(const float* __restrict__ Y, float* ws) {
  int wave = threadIdx.x >> 5, lane = threadIdx.x & 31;
  int half = lane >> 4, r = lane & 15;
  int gwid = blockIdx.x * 8 + wave;
  int row0 = (gwid >> 6) * 16, col0 = (gwid & 63) * 16;
  const float* y2 = ws + OFF_Y2;
  float* SQ = ws + OFF_SQ;
  v8f c = {};
  for (int s = 0; s < 16; ++s) {
    int k0 = 4 * s + 2 * half;
    v2f a, b;
    a[0] = Y[(row0 + r) * 64 + k0]; a[1] = Y[(row0 + r) * 64 + k0 + 1];
    b[0] = Y[(col0 + r) * 64 + k0]; b[1] = Y[(col0 + r) * 64 + k0 + 1];
    c = wmma_f32(a, b, c);
  }
  float y2c = y2[col0 + r];
  float lmin = 3.0e38f;
  for (int e = 0; e < 8; ++e) {
    int grow = row0 + e + 8 * half, gcol = col0 + r;
    float d = fmaxf(y2[grow] + y2c - 2.f * c[e], 0.f);
    SQ[grow * 1024 + gcol] = d;
    float dd = (grow == gcol) ? 100.f : d;
    lmin = fminf(lmin, dd);
  }
  for (int m = 16; m; m >>= 1) lmin = fminf(lmin, __shfl_xor(lmin, m, 32));
  if (lane == 0) atomicMin((unsigned*)(ws + OFF_ACC) + 0, __float_as_uint(lmin));
}

__global__ void k_graph_rowsum(float* ws) {
  __shared__ float red[8];
  int i = blockIdx.x;
  const float* SQ = ws + OFF_SQ;
  float gy = __uint_as_float(((unsigned*)(ws + OFF_ACC))[0]);
  float s = 0.f;
  for (int k = 0; k < 4; ++k) {
    int j = threadIdx.x + k * 256;
    float val = SQ[i * 1024 + j];
    if (j == i) val = 100.f;
    s += __expf(-ALPHA * fminf(val + gy, 10.f));
  }
  for (int m = 16; m; m >>= 1) s += __shfl_xor(s, m, 32);
  int lane = threadIdx.x & 31, w = threadIdx.x >> 5;
  if (lane == 0) red[w] = s;
  __syncthreads();
  if (threadIdx.x == 0) {
    float t = 0.f;
    for (int w2 = 0; w2 < 8; ++w2) t += red[w2];
    ws[OFF_ROWSUM + i] = t;
  }
}

// A_ij = 0.5*f(sq_ij)*(1/rs_i + 1/rs_j)  (sq symmetric => E_ji = f(sq_ij))
// fused: deg_i and stretch (j>i)
__global__ void k_graph_A(float* ws) {
  __shared__ float redd[8], reds[8];
  int i = blockIdx.x;
  const float* SQ = ws + OFF_SQ;
  const float* RS = ws + OFF_ROWSUM;
  float* A = ws + OFF_A;
  float gy = __uint_as_float(((unsigned*)(ws + OFF_ACC))[0]);
  float inv_i = 1.f / RS[i];
  float deg = 0.f, st = 0.f;
  for (int k = 0; k < 4; ++k) {
    int j = threadIdx.x + k * 256;
    float sq = SQ[i * 1024 + j];
    float val = (j == i) ? 100.f : sq;
    float f = __expf(-ALPHA * fminf(val + gy, 10.f));
    float aij = 0.5f * f * (inv_i + 1.f / RS[j]);
    A[i * 1024 + j] = aij;
    deg += aij;
    if (j > i) st += sq * aij;
  }
  for (int m = 16; m; m >>= 1) { deg += __shfl_xor(deg, m, 32); st += __shfl_xor(st, m, 32); }
  int lane = threadIdx.x & 31, w = threadIdx.x >> 5;
  if (lane == 0) { redd[w] = deg; reds[w] = st; }
  __syncthreads();
  if (threadIdx.x == 0) {
    float td = 0.f, ts = 0.f;
    for (int w2 = 0; w2 < 8; ++w2) { td += redd[w2]; ts += reds[w2]; }
    ws[OFF_DEG + i] = td;
    atomicAdd(ws + OFF_ACC + 3, ts);
  }
}

// AY = A @ y via WMMA, K=1024 (256 k-steps)
__global__ void k_Ay(const float* __restrict__ Y, float* ws) {
  int wave = threadIdx.x >> 5, lane = threadIdx.x & 31;
  int half = lane >> 4, r = lane & 15;
  int gwid = blockIdx.x * 8 + wave;           // 256 tiles: 64 row x 4 col
  int row0 = (gwid >> 2) * 16, col0 = (gwid & 3) * 16;
  const float* A = ws + OFF_A;
  float* AY = ws + OFF_AY;
  v8f c = {};
  for (int s = 0; s < 256; ++s) {
    int k0 = 4 * s + 2 * half;
    v2f a, b;
    a[0] = A[(row0 + r) * 1024 + k0]; a[1] = A[(row0 + r) * 1024 + k0 + 1];
    b[0] = Y[k0 * 64 + col0 + r];     b[1] = Y[(k0 + 1) * 64 + col0 + r];
    c = wmma_f32(a, b, c);
  }
  for (int e = 0; e < 8; ++e)
    AY[(row0 + e + 8 * half) * 64 + col0 + r] = c[e];
}

__global__ void k_harm(const float* __restrict__ Y, float* ws) {
  int wave = threadIdx.x >> 5, lane = threadIdx.x & 31;
  int row = blockIdx.x * 8 + wave;
  float deg = ws[OFF_DEG + row];
  if (deg > 1.f) {
    const float* AY = ws + OFF_AY;
    float inv = 1.f / deg;
    int i0 = row * 64 + lane;
    float d0 = Y[i0] - AY[i0] * inv;
    float d1 = Y[i0 + 32] - AY[i0 + 32] * inv;
    float v = d0 * d0 + d1 * d1;
    for (int m = 16; m; m >>= 1) v += __shfl_xor(v, m, 32);
    if (lane == 0) atomicAdd(ws + OFF_ACC + 4, v);
  }
}

// pass 1 over X x y^T: global min of dst (bits-ordered atomicMin, d>=0)
__global__ void k_big_min(const float* __restrict__ X, const float* __restrict__ Y, float* ws) {
  int wave = threadIdx.x >> 5, lane = threadIdx.x & 31;
  int half = lane >> 4, r = lane & 15;
  int gwid = blockIdx.x * 8 + wave;           // 6250*64 tiles
  int row0 = (gwid >> 6) * 16, col0 = (gwid & 63) * 16;
  const float* x2 = ws + OFF_X2;
  const float* y2 = ws + OFF_Y2;
  v8f c = {};
  for (int s = 0; s < 16; ++s) {
    int k0 = 4 * s + 2 * half;
    v2f a, b;
    a[0] = X[(row0 + r) * 64 + k0]; a[1] = X[(row0 + r) * 64 + k0 + 1];
    b[0] = Y[(col0 + r) * 64 + k0]; b[1] = Y[(col0 + r) * 64 + k0 + 1];
    c = wmma_f32(a, b, c);
  }
  float y2c = y2[col0 + r];
  float lmin = 3.0e38f;
  for (int e = 0; e < 8; ++e) {
    float d = fmaxf(x2[row0 + e + 8 * half] + y2c - 2.f * c[e], 0.f);
    lmin = fminf(lmin, d);
  }
  for (int m = 16; m; m >>= 1) lmin = fminf(lmin, __shfl_xor(lmin, m, 32));
  if (lane == 0) atomicMin((unsigned*)(ws + OFF_ACC) + 1, __float_as_uint(lmin));
}

// pass 2: 16-row block per WG; wave w covers col tiles w,w+8,...; fused softmax
// accumulators S0=sum e, S1=sum d*e per row -> mse_i = S1/S0 (no dst storage)
__global__ void k_big_mse(const float* __restrict__ X, const float* __restrict__ Y, float* ws) {
  __shared__ float ls0[16], ls1[16];
  int wave = threadIdx.x >> 5, lane = threadIdx.x & 31;
  int half = lane >> 4, r = lane & 15;
  int row0 = blockIdx.x * 16;
  const float* x2 = ws + OFF_X2;
  const float* y2 = ws + OFF_Y2;
  float gmin = __uint_as_float(((unsigned*)(ws + OFF_ACC))[1]);
  if (threadIdx.x < 16) { ls0[threadIdx.x] = 0.f; ls1[threadIdx.x] = 0.f; }
  __syncthreads();
  v2f afr[16];
  for (int s = 0; s < 16; ++s) {
    int k0 = 4 * s + 2 * half;
    afr[s][0] = X[(row0 + r) * 64 + k0];
    afr[s][1] = X[(row0 + r) * 64 + k0 + 1];
  }
  float x2v[8];
  for (int e = 0; e < 8; ++e) x2v[e] = x2[row0 + e + 8 * half];
  float s0[8], s1[8];
  for (int e = 0; e < 8; ++e) { s0[e] = 0.f; s1[e] = 0.f; }
  for (int t = 0; t < 8; ++t) {
    int col0 = (wave + t * 8) * 16;
    v8f c = {};
    for (int s = 0; s < 16; ++s) {
      int k0 = 4 * s + 2 * half;
      v2f b;
      b[0] = Y[(col0 + r) * 64 + k0];
      b[1] = Y[(col0 + r) * 64 + k0 + 1];
      c = wmma_f32(afr[s], b, c);
    }
    float y2c = y2[col0 + r];
    for (int e = 0; e < 8; ++e) {
      float d = fmaxf(x2v[e] + y2c - 2.f * c[e], 0.f);
      float ee = __expf(-ALPHA * fminf(d + gmin, 10.f));
      s0[e] += ee;
      s1[e] += d * ee;
    }
  }
  // reduce across the 16 lanes sharing a row (xor masks stay inside each half)
  for (int e = 0; e < 8; ++e)
    for (int m = 1; m < 16; m <<= 1) {
      s0[e] += __shfl_xor(s0[e], m, 32);
      s1[e] += __shfl_xor(s1[e], m, 32);
    }
  if (r == 0) {
    for (int e = 0; e < 8; ++e) {
      atomicAdd(&ls0[e + 8 * half], s0[e]);
      atomicAdd(&ls1[e + 8 * half], s1[e]);
    }
  }
  __syncthreads();
  float contrib = (threadIdx.x < 16) ? (ls1[threadIdx.x] / ls0[threadIdx.x]) : 0.f;
  if (wave == 0) {
    for (int m = 16; m; m >>= 1) contrib += __shfl_xor(contrib, m, 32);
    if (lane == 0) atomicAdd(ws + OFF_ACC + 2, contrib);
  }
}

__global__ void k_finalize(float* ws, float* out) {
  if (threadIdx.x == 0) {
    float loss = ws[OFF_ACC + 2] / (float)N_X
               + LMDA * ws[OFF_ACC + 3]
               + MU * ws[OFF_ACC + 4];
    out[M_Y * 64] = loss;
  }
}

__global__ void k_copy_y(const float* __restrict__ Y, float* out) {
  int i = blockIdx.x * 256 + threadIdx.x;
  out[i] = Y[i];
}

extern "C" void kernel_launch(void* const* d_in, const int* in_sizes, int n_in,
                              void* d_out, int out_size, void* d_ws, size_t ws_size,
                              hipStream_t stream) {
  (void)in_sizes; (void)n_in; (void)out_size; (void)ws_size;
  const float* X = (const float*)d_in[0];
  const float* Y = (const float*)d_in[1];
  float* out = (float*)d_out;
  float* ws = (float*)d_ws;

  k_init_acc   <<<1, 32, 0, stream>>>(ws);
  k_rownorm    <<<(N_X + 7) / 8, 256, 0, stream>>>(X, ws + OFF_X2, N_X);
  k_rownorm    <<<M_Y / 8, 256, 0, stream>>>(Y, ws + OFF_Y2, M_Y);
  k_graph_dst  <<<(64 * 64) / 8, 256, 0, stream>>>(Y, ws);
  k_graph_rowsum<<<1024, 256, 0, stream>>>(ws);
  k_graph_A    <<<1024, 256, 0, stream>>>(ws);
  k_Ay         <<<256 / 8, 256, 0, stream>>>(Y, ws);
  k_harm       <<<1024 / 8, 256, 0, stream>>>(Y, ws);
  k_big_min    <<<(6250 * 64) / 8, 256, 0, stream>>>(X, Y, ws);
  k_big_mse    <<<6250, 256, 0, stream>>>(X, Y, ws);
  k_finalize   <<<1, 32, 0, stream>>>(ws, out);
  k_copy_y     <<<(M_Y * 64) / 256, 256, 0, stream>>>(Y, out);
}